// SCG_1022202216906
// MI455X (gfx1250) — compile-verified
//
#include <hip/hip_runtime.h>
#include <math.h>

typedef __attribute__((ext_vector_type(2))) float v2f;
typedef __attribute__((ext_vector_type(8))) float v8f;

#define WMMA_F32_4(a, b, c) \
  __builtin_amdgcn_wmma_f32_16x16x4_f32(false, (a), false, (b), (short)0, (c), false, false)

// Problem constants: B=16, C=1024, H=W=64, nodes 32x32, N=1024, HID=16

// ---------------------------------------------------------------------------
// K0: 2x2 average pool, feature (16,1024,64,64) -> nodes (16,1024,32,32)
// ---------------------------------------------------------------------------
__global__ __launch_bounds__(256) void k_pool(const float* __restrict__ f,
                                              float* __restrict__ nodes) {
  unsigned idx = blockIdx.x * 256u + threadIdx.x;   // < 8,388,608
  unsigned t   = idx & 15u;                         // x-pair id (2 outputs)
  unsigned row = idx >> 4;
  unsigned ny  = row & 31u;
  unsigned bc  = row >> 5;                          // < 16384
  const float* src = f + (size_t)bc * 4096 + ny * 128 + t * 4;
  float4 r0 = *(const float4*)(src);
  float4 r1 = *(const float4*)(src + 64);
  float2 o  = make_float2((r0.x + r0.y + r1.x + r1.y) * 0.25f,
                          (r0.z + r0.w + r1.z + r1.w) * 0.25f);
  *(float2*)(nodes + (size_t)bc * 1024 + ny * 32 + t * 2) = o;
}

// ---------------------------------------------------------------------------
// K1: fused conv3x3 (SAME) + conv1x1 via implicit GEMM with f32 WMMA.
// One BLOCK per 16(spatial)x16(oc) output tile; 8 waves split the 1024-channel
// K range (128 channels each, 10 taps). Each wave stages only its own weight
// slice in LDS (disjoint -> no barriers in the K loop). LDS tree-reduce + bias.
// ---------------------------------------------------------------------------
__global__ __launch_bounds__(256) void k_conv(const float* __restrict__ nodes,
                                              const float* __restrict__ muw,
                                              const float* __restrict__ mub,
                                              const float* __restrict__ lvw,
                                              const float* __restrict__ lvb,
                                              float* __restrict__ mu,
                                              float* __restrict__ lv) {
  __shared__ float wlds[16384];                 // per-wave slices [w][c_local][oc]
  int tid = threadIdx.x;
  int wv = tid >> 5, lane = tid & 31;
  int tile = blockIdx.x;                        // 0..1023 = B*32*2
  int b = tile >> 6;
  int rem = tile & 63;
  int y = rem >> 1;
  int x0 = (rem & 1) * 16;
  int m = lane & 15;
  bool hi = lane >= 16;
  int kh = hi ? 2 : 0;                          // K-half offset for A/B operands
  int cbase = wv * 128;                         // this wave's channel slice
  float* myw = wlds + wv * 2048;                // [c_local(128)][oc(16)]
  const float* nb = nodes + (size_t)b * 1048576;

  v8f cmu = {};
  v8f clv = {};

  // ---- taps 0..8 accumulate mu ----
  for (int tap = 0; tap < 9; ++tap) {
    for (int i = lane; i < 2048; i += 32) {     // stage my weight slice
      int c = cbase + (i >> 4), oc = i & 15;
      myw[i] = muw[(size_t)oc * 9216 + c * 9 + tap];
    }
    int dy = tap / 3 - 1, dx = tap % 3 - 1;
    int yy = y + dy;
    int xx = x0 + m + dx;
    bool valid = (yy >= 0) && (yy < 32) && (xx >= 0) && (xx < 32);
    const float* np = nb + min(max(yy, 0), 31) * 32 + min(max(xx, 0), 31);
    for (int cl = 0; cl < 128; cl += 4) {
      int ck = cbase + cl + kh;
      float ax = np[ck * 1024];
      float ay = np[ck * 1024 + 1024];
      v2f a  = {valid ? ax : 0.f, valid ? ay : 0.f};
      v2f bb = {myw[(cl + kh) * 16 + m], myw[(cl + kh + 1) * 16 + m]};
      cmu = WMMA_F32_4(a, bb, cmu);
    }
  }

  // ---- tap 9: 1x1 conv accumulates log_var ----
  {
    for (int i = lane; i < 2048; i += 32) {
      int c = cbase + (i >> 4), oc = i & 15;
      myw[i] = lvw[oc * 1024 + c];
    }
    const float* np = nb + y * 32 + x0 + m;     // center tap always valid
    for (int cl = 0; cl < 128; cl += 4) {
      int ck = cbase + cl + kh;
      v2f a  = {np[ck * 1024], np[ck * 1024 + 1024]};
      v2f bb = {myw[(cl + kh) * 16 + m], myw[(cl + kh + 1) * 16 + m]};
      clv = WMMA_F32_4(a, bb, clv);
    }
  }

  // ---- cross-wave reduction of partial tiles ----
  __syncthreads();                              // all waves done with weight LDS
#pragma unroll
  for (int v = 0; v < 8; ++v) {
    wlds[wv * 256 + lane * 8 + v]        = cmu[v];
    wlds[2048 + wv * 256 + lane * 8 + v] = clv[v];
  }
  __syncthreads();
  float sm = 0.f, sl = 0.f;
#pragma unroll
  for (int w = 0; w < 8; ++w) {
    sm += wlds[w * 256 + tid];
    sl += wlds[2048 + w * 256 + tid];
  }
  // element id: lane_e = tid>>3 , v_e = tid&7 ; D layout (M,N)
  int lane_e = tid >> 3, v_e = tid & 7;
  int oc = lane_e & 15;
  int sp = x0 + v_e + (lane_e >= 16 ? 8 : 0);
  size_t addr = (size_t)b * 16384 + oc * 1024 + y * 32 + sp;
  mu[addr] = sm + mub[oc];
  lv[addr] = sl + lvb[oc];
}

// ---------------------------------------------------------------------------
// K2: per-node stats. Ad=||z||^2, sigma_b=sum Ad, slog=sum log(Ad+1e-7),
// kl partial = sum(1 + 2*lv - mu^2 - exp(2*lv)).
// ---------------------------------------------------------------------------
__global__ __launch_bounds__(256) void k_stats(const float* __restrict__ mu,
                                               const float* __restrict__ lv,
                                               float* __restrict__ sigma,
                                               float* __restrict__ klsum,
                                               float* __restrict__ slog) {
  __shared__ float s0[256], s1[256], s2[256];
  int i = blockIdx.x * 256 + threadIdx.x;       // node id < 16384
  const float4* mp = (const float4*)(mu + (size_t)i * 16);
  const float4* lp = (const float4*)(lv + (size_t)i * 16);
  float ad = 0.f, kl = 0.f;
  for (int q = 0; q < 4; ++q) {
    float4 mv = mp[q], lw = lp[q];
    float sq = mv.x * mv.x + mv.y * mv.y + mv.z * mv.z + mv.w * mv.w;
    ad += sq;
    kl += 4.f + 2.f * (lw.x + lw.y + lw.z + lw.w) - sq -
          (expf(2.f * lw.x) + expf(2.f * lw.y) + expf(2.f * lw.z) + expf(2.f * lw.w));
  }
  s0[threadIdx.x] = ad;
  s1[threadIdx.x] = kl;
  s2[threadIdx.x] = logf(ad + 1e-7f);
  __syncthreads();
  for (int s = 128; s > 0; s >>= 1) {
    if (threadIdx.x < (unsigned)s) {
      s0[threadIdx.x] += s0[threadIdx.x + s];
      s1[threadIdx.x] += s1[threadIdx.x + s];
      s2[threadIdx.x] += s2[threadIdx.x + s];
    }
    __syncthreads();
  }
  if (threadIdx.x == 0) {
    atomicAdd(&sigma[blockIdx.x >> 2], s0[0]);
    atomicAdd(klsum, s1[0]);
    atomicAdd(slog, s2[0]);
  }
}

// ---------------------------------------------------------------------------
// K3: gamma_b, dl_loss (B,1,1), kl_loss scalar
// ---------------------------------------------------------------------------
__global__ void k_final(const float* __restrict__ sigma,
                        const float* __restrict__ klsum,
                        const float* __restrict__ slog,
                        float* __restrict__ gamma,
                        float* __restrict__ okl, float* __restrict__ odl) {
  int b = threadIdx.x;
  if (b < 16) {
    float g = sqrtf(1.f + 1024.f / (sigma[b] + 1e-7f));
    gamma[b] = g;
    odl[b] = -g * (*slog) * (1.f / (16.f * 1024.f * 1024.f));
  }
  if (b == 0) *okl = -0.5f * (*klsum);
}

// ---------------------------------------------------------------------------
// K4: z = mu (raw reshape == flat copy), z_hat = gamma * mu * (1 - lv)
// ---------------------------------------------------------------------------
__global__ __launch_bounds__(256) void k_zh(const float* __restrict__ mu,
                                            const float* __restrict__ lv,
                                            const float* __restrict__ gamma,
                                            float* __restrict__ z,
                                            float* __restrict__ zh) {
  int i = blockIdx.x * 256 + threadIdx.x;       // < 262144
  float m = mu[i], l = lv[i];
  z[i]  = m;
  zh[i] = gamma[i >> 14] * m * (1.f - l);
}

// ---------------------------------------------------------------------------
// One 16x16 Gram tile A[I,J] = Z_I * Z_J^T via 4x WMMA f32 16x16x4.
// Lane loads float2 directly at its K-half offset (kh) -> no selects.
// ---------------------------------------------------------------------------
__device__ inline v8f gram_tile(const float* __restrict__ Zb, int I, int J,
                                int r, int kh) {
  const float* pA = Zb + (size_t)(I * 16 + r) * 16 + kh;
  const float* pB = Zb + (size_t)(J * 16 + r) * 16 + kh;
  v8f c = {};
#pragma unroll
  for (int kk = 0; kk < 4; ++kk) {
    v2f a = *(const v2f*)(pA + kk * 4);
    v2f b = *(const v2f*)(pB + kk * 4);
    c = WMMA_F32_4(a, b, c);
  }
  return c;
}

// ---------------------------------------------------------------------------
// K5: pass 1 — column (axis=1) degree sums of A (relu + diag mods + eye)
// ---------------------------------------------------------------------------
__global__ __launch_bounds__(256) void k_deg(const float* __restrict__ Z,
                                             const float* __restrict__ gamma,
                                             float* __restrict__ deg) {
  int wv = threadIdx.x >> 5, lane = threadIdx.x & 31;
  int tile = blockIdx.x * 8 + wv;               // < 65536
  int b = tile >> 12;
  int rem = tile & 4095;
  int I = rem >> 6, J = rem & 63;
  int r = lane & 15;
  bool hi = lane >= 16;
  v8f c = gram_tile(Z + (size_t)b * 16384, I, J, r, hi ? 2 : 0);
  float g = gamma[b];
  int col = J * 16 + r;
  int rbase = I * 16 + (hi ? 8 : 0);
  float cs = 0.f;
#pragma unroll
  for (int v = 0; v < 8; ++v) {
    int row = rbase + v;
    float val = fmaxf(c[v], 0.f);
    if (row == col) val = val * (1.f + g) + 1.f;   // + gamma*Ad*eye + eye
    cs += val;
  }
  cs += __shfl_xor(cs, 16);                     // combine row halves per column
  if (!hi) atomicAdd(&deg[b * 1024 + col], cs);
}

__global__ __launch_bounds__(256) void k_dinv(float* __restrict__ deg) {
  int i = blockIdx.x * 256 + threadIdx.x;       // < 16384
  deg[i] = 1.0f / sqrtf(deg[i] + 1e-5f);
}

// ---------------------------------------------------------------------------
// K6: pass 2 — recompute tile, normalize, write LA (coalesced along columns)
// ---------------------------------------------------------------------------
__global__ __launch_bounds__(256) void k_la(const float* __restrict__ Z,
                                            const float* __restrict__ gamma,
                                            const float* __restrict__ dinv,
                                            float* __restrict__ LA) {
  int wv = threadIdx.x >> 5, lane = threadIdx.x & 31;
  int tile = blockIdx.x * 8 + wv;
  int b = tile >> 12;
  int rem = tile & 4095;
  int I = rem >> 6, J = rem & 63;
  int r = lane & 15;
  bool hi = lane >= 16;
  v8f c = gram_tile(Z + (size_t)b * 16384, I, J, r, hi ? 2 : 0);
  float g = gamma[b];
  int col = J * 16 + r;
  float dc = dinv[b * 1024 + col];
  int rbase = I * 16 + (hi ? 8 : 0);
  float* out = LA + (size_t)b * 1048576 + col;
#pragma unroll
  for (int v = 0; v < 8; ++v) {
    int row = rbase + v;
    float val = fmaxf(c[v], 0.f);
    if (row == col) val = val * (1.f + g) + 1.f;
    val *= dinv[b * 1024 + row] * dc;
    out[(size_t)row * 1024] = val;
  }
}

// ---------------------------------------------------------------------------
extern "C" void kernel_launch(void* const* d_in, const int* in_sizes, int n_in,
                              void* d_out, int out_size, void* d_ws, size_t ws_size,
                              hipStream_t stream) {
  const float* feature = (const float*)d_in[0];
  const float* muw = (const float*)d_in[1];
  const float* mub = (const float*)d_in[2];
  const float* lvw = (const float*)d_in[3];
  const float* lvb = (const float*)d_in[4];

  float* ws    = (float*)d_ws;
  float* nodes = ws;                      // 16,777,216 f (64 MB)
  float* mu    = ws + 16777216;           // 262,144 f
  float* lv    = mu + 262144;             // 262,144 f
  float* deg   = lv + 262144;             // 16,384 f
  float* sigma = deg + 16384;             // 16 f
  float* gamma = sigma + 16;              // 16 f
  float* klsum = gamma + 16;              // 1 f
  float* slog  = klsum + 1;               // 1 f

  float* out = (float*)d_out;
  float* oLA = out;                       // 16,777,216
  float* oZ  = out + 16777216;            // 262,144
  float* oZH = oZ + 262144;               // 262,144
  float* oKL = oZH + 262144;              // 1
  float* oDL = oKL + 1;                   // 16

  hipMemsetAsync(deg, 0, (16384 + 16 + 16 + 2) * sizeof(float), stream);

  k_pool <<<32768, 256, 0, stream>>>(feature, nodes);
  k_conv <<< 1024, 256, 0, stream>>>(nodes, muw, mub, lvw, lvb, mu, lv);
  k_stats<<<   64, 256, 0, stream>>>(mu, lv, sigma, klsum, slog);
  k_final<<<    1,  32, 0, stream>>>(sigma, klsum, slog, gamma, oKL, oDL);
  k_zh   <<< 1024, 256, 0, stream>>>(mu, lv, gamma, oZ, oZH);
  k_deg  <<< 8192, 256, 0, stream>>>(mu, gamma, deg);
  k_dinv <<<   64, 256, 0, stream>>>(deg);
  k_la   <<< 8192, 256, 0, stream>>>(mu, gamma, deg, oLA);
}